// GCN_Node_53824530153630
// MI455X (gfx1250) — compile-verified
//
#include <hip/hip_runtime.h>
#include <hip/hip_bf16.h>

typedef __attribute__((ext_vector_type(2))) float v2f;
typedef __attribute__((ext_vector_type(8))) float v8f;

#define FDIM 64
#define NCLS 10

// ---------------- degree / normalization ----------------

__global__ __launch_bounds__(256) void k_deg_init(float* __restrict__ deg, int n) {
  int i = blockIdx.x * blockDim.x + threadIdx.x;
  if (i < n) deg[i] = 1.0f;  // self-loop contributes 1
}

__global__ __launch_bounds__(256) void k_deg_accum(const long long* __restrict__ dst,
                                                   float* __restrict__ deg, int E) {
  int e = blockIdx.x * blockDim.x + threadIdx.x;
  if (e < E) atomicAdd(&deg[(int)dst[e]], 1.0f);
}

__global__ __launch_bounds__(256) void k_deg_rsqrt(float* __restrict__ deg, int n) {
  int i = blockIdx.x * blockDim.x + threadIdx.x;
  if (i < n) deg[i] = rsqrtf(deg[i]);
}

// ---------------- WMMA fp32 GEMM: Y[n,64] = X[n,64] @ W[64,64] (+bias,relu) ----------------
// 128 threads = 4 waves; each wave computes a 16x64 output tile via
// V_WMMA_F32_16X16X4_F32 over 16 K-steps, 4 N-tiles. W staged in LDS.

__global__ __launch_bounds__(128) void k_gemm64_wmma(const float* __restrict__ X,
                                                     const float* __restrict__ W,
                                                     const float* __restrict__ bias,
                                                     float* __restrict__ Y,
                                                     int n, int fuse_bias_relu) {
  __shared__ float sW[FDIM * FDIM];
  for (int i = threadIdx.x; i < FDIM * FDIM; i += 128) sW[i] = W[i];
  __syncthreads();

  const int wave = threadIdx.x >> 5;
  const int lane = threadIdx.x & 31;
  const int half = lane >> 4;   // 0: K/M low half, 1: high half (per ISA layout)
  const int l    = lane & 15;
  const int rowBase = (blockIdx.x * 4 + wave) * 16;

  int arow = rowBase + l;
  if (arow >= n) arow = n - 1;              // clamp; padded rows never stored
  const float* __restrict__ Ar = X + (size_t)arow * FDIM;

  v8f acc0 = {}; v8f acc1 = {}; v8f acc2 = {}; v8f acc3 = {};
  #pragma unroll
  for (int k = 0; k < FDIM; k += 4) {
    const int ka = k + half * 2;
    v2f a; a.x = Ar[ka]; a.y = Ar[ka + 1];
    const float* w0 = &sW[ka * FDIM + l];
    const float* w1 = &sW[(ka + 1) * FDIM + l];
    v2f b0, b1, b2, b3;
    b0.x = w0[0];  b0.y = w1[0];
    b1.x = w0[16]; b1.y = w1[16];
    b2.x = w0[32]; b2.y = w1[32];
    b3.x = w0[48]; b3.y = w1[48];
    acc0 = __builtin_amdgcn_wmma_f32_16x16x4_f32(false, a, false, b0, (short)0, acc0, false, false);
    acc1 = __builtin_amdgcn_wmma_f32_16x16x4_f32(false, a, false, b1, (short)0, acc1, false, false);
    acc2 = __builtin_amdgcn_wmma_f32_16x16x4_f32(false, a, false, b2, (short)0, acc2, false, false);
    acc3 = __builtin_amdgcn_wmma_f32_16x16x4_f32(false, a, false, b3, (short)0, acc3, false, false);
  }

  #pragma unroll
  for (int r = 0; r < 8; ++r) {
    const int m = rowBase + r + half * 8;   // C/D layout: M = r + 8*(lane>=16)
    if (m < n) {
      float v0 = acc0[r], v1 = acc1[r], v2 = acc2[r], v3 = acc3[r];
      if (fuse_bias_relu) {
        v0 = fmaxf(v0 + bias[0  + l], 0.0f);
        v1 = fmaxf(v1 + bias[16 + l], 0.0f);
        v2 = fmaxf(v2 + bias[32 + l], 0.0f);
        v3 = fmaxf(v3 + bias[48 + l], 0.0f);
      }
      float* yr = Y + (size_t)m * FDIM;
      yr[0  + l] = v0;
      yr[16 + l] = v1;
      yr[32 + l] = v2;
      yr[48 + l] = v3;
    }
  }
}

// ---------------- aggregation ----------------

// seed with self-loop: out[i,f] = dinv[i]^2 * h[i,f]
__global__ __launch_bounds__(256) void k_agg_seed(const float* __restrict__ h,
                                                  const float* __restrict__ dinv,
                                                  float* __restrict__ out, int n) {
  long long idx = (long long)blockIdx.x * blockDim.x + threadIdx.x;
  if (idx < (long long)n * FDIM) {
    int i = (int)(idx >> 6);
    float di = dinv[i];
    out[idx] = di * di * h[idx];
  }
}

// scatter: out[dst,f] += dinv[src]*dinv[dst]*h[src,f]; one thread per (edge,feature)
__global__ __launch_bounds__(256) void k_agg_edges(const long long* __restrict__ src,
                                                   const long long* __restrict__ dst,
                                                   const float* __restrict__ dinv,
                                                   const float* __restrict__ h,
                                                   float* __restrict__ out, int E) {
  long long t = (long long)blockIdx.x * blockDim.x + threadIdx.x;
  long long e = t >> 6;
  int f = (int)(t & 63);
  if (e >= E) return;
  int s = (int)src[e];
  int d = (int)dst[e];
  float w = dinv[s] * dinv[d];
  atomicAdd(&out[(size_t)d * FDIM + f], w * h[(size_t)s * FDIM + f]);
}

// in-place x = relu(x + b)
__global__ __launch_bounds__(256) void k_bias_relu(float* __restrict__ x,
                                                   const float* __restrict__ b, int n) {
  long long idx = (long long)blockIdx.x * blockDim.x + threadIdx.x;
  if (idx < (long long)n * FDIM) {
    x[idx] = fmaxf(x[idx] + b[idx & 63], 0.0f);
  }
}

// ---------------- head: logits = h@Wl2 + bl2 ; log_softmax ----------------

__global__ __launch_bounds__(256) void k_head(const float* __restrict__ h,
                                              const float* __restrict__ Wl2,
                                              const float* __restrict__ bl2,
                                              float* __restrict__ out, int n) {
  __shared__ float sW[FDIM * NCLS];
  __shared__ float sb[NCLS];
  for (int i = threadIdx.x; i < FDIM * NCLS; i += 256) sW[i] = Wl2[i];
  if (threadIdx.x < NCLS) sb[threadIdx.x] = bl2[threadIdx.x];
  __syncthreads();

  int i = blockIdx.x * blockDim.x + threadIdx.x;
  if (i >= n) return;
  const float* hr = h + (size_t)i * FDIM;
  float logits[NCLS];
  #pragma unroll
  for (int c = 0; c < NCLS; ++c) logits[c] = sb[c];
  #pragma unroll 4
  for (int k = 0; k < FDIM; ++k) {
    float hv = hr[k];
    #pragma unroll
    for (int c = 0; c < NCLS; ++c) logits[c] += hv * sW[k * NCLS + c];
  }
  float m = logits[0];
  #pragma unroll
  for (int c = 1; c < NCLS; ++c) m = fmaxf(m, logits[c]);
  float s = 0.0f;
  #pragma unroll
  for (int c = 0; c < NCLS; ++c) s += __expf(logits[c] - m);
  float lse = m + __logf(s);
  float* orow = out + (size_t)i * NCLS;
  #pragma unroll
  for (int c = 0; c < NCLS; ++c) orow[c] = logits[c] - lse;
}

// ---------------- launcher ----------------

extern "C" void kernel_launch(void* const* d_in, const int* in_sizes, int n_in,
                              void* d_out, int out_size, void* d_ws, size_t ws_size,
                              hipStream_t stream) {
  const float*     x   = (const float*)d_in[0];
  const long long* ei  = (const long long*)d_in[1];   // int64 per reference
  const float*     W1  = (const float*)d_in[2];
  const float*     b1  = (const float*)d_in[3];
  const float*     W2  = (const float*)d_in[4];
  const float*     b2  = (const float*)d_in[5];
  const float*     W3  = (const float*)d_in[6];
  const float*     b3  = (const float*)d_in[7];
  const float*     Wl1 = (const float*)d_in[8];
  const float*     bl1 = (const float*)d_in[9];
  const float*     Wl2 = (const float*)d_in[10];
  const float*     bl2 = (const float*)d_in[11];

  const int n = in_sizes[0] / FDIM;   // 100000
  const int E = in_sizes[1] / 2;      // 1600000
  const long long* srcIdx = ei;       // edge_index[0]
  const long long* dstIdx = ei + E;   // edge_index[1]

  float* dinv = (float*)d_ws;                   // [n]
  float* bufA = dinv + n;                       // [n,64]
  float* bufB = bufA + (size_t)n * FDIM;        // [n,64]
  float* outp = (float*)d_out;                  // [n,10]

  const int TB = 256;
  dim3 blk256(TB);
  dim3 gN((n + TB - 1) / TB);
  dim3 gE((E + TB - 1) / TB);
  dim3 gNF((int)(((long long)n * FDIM + TB - 1) / TB));
  dim3 gEF((int)(((long long)E * FDIM + TB - 1) / TB));
  dim3 gGemm((n + 63) / 64);
  dim3 blkGemm(128);

  // normalization: deg (incl. self loop) -> dinv = rsqrt(deg)
  k_deg_init <<<gN, blk256, 0, stream>>>(dinv, n);
  k_deg_accum<<<gE, blk256, 0, stream>>>(dstIdx, dinv, E);
  k_deg_rsqrt<<<gN, blk256, 0, stream>>>(dinv, n);

  // conv layer 1: input x
  k_gemm64_wmma<<<gGemm, blkGemm, 0, stream>>>(x, W1, nullptr, bufB, n, 0);
  k_agg_seed   <<<gNF, blk256, 0, stream>>>(bufB, dinv, bufA, n);
  k_agg_edges  <<<gEF, blk256, 0, stream>>>(srcIdx, dstIdx, dinv, bufB, bufA, E);
  k_bias_relu  <<<gNF, blk256, 0, stream>>>(bufA, b1, n);

  // conv layer 2
  k_gemm64_wmma<<<gGemm, blkGemm, 0, stream>>>(bufA, W2, nullptr, bufB, n, 0);
  k_agg_seed   <<<gNF, blk256, 0, stream>>>(bufB, dinv, bufA, n);
  k_agg_edges  <<<gEF, blk256, 0, stream>>>(srcIdx, dstIdx, dinv, bufB, bufA, E);
  k_bias_relu  <<<gNF, blk256, 0, stream>>>(bufA, b2, n);

  // conv layer 3
  k_gemm64_wmma<<<gGemm, blkGemm, 0, stream>>>(bufA, W3, nullptr, bufB, n, 0);
  k_agg_seed   <<<gNF, blk256, 0, stream>>>(bufB, dinv, bufA, n);
  k_agg_edges  <<<gEF, blk256, 0, stream>>>(srcIdx, dstIdx, dinv, bufB, bufA, E);
  k_bias_relu  <<<gNF, blk256, 0, stream>>>(bufA, b3, n);

  // linear 1 with fused bias+relu
  k_gemm64_wmma<<<gGemm, blkGemm, 0, stream>>>(bufA, Wl1, bl1, bufB, n, 1);

  // head + log_softmax
  k_head<<<gN, blk256, 0, stream>>>(bufB, Wl2, bl2, outp, n);
}